// SubglacialDrainageSystem_11115375362948
// MI455X (gfx1250) — compile-verified
//
#include <hip/hip_runtime.h>
#include <hip/hip_bf16.h>
#include <stdint.h>

// ---------------- physics constants (match reference) ----------------
#define K_SHEET   0.01f
#define K_CHAN    0.1f
#define CAV_SPACE 2.0f
// DISS = (1/917 - 1/1000)/3.34e5
__device__ __constant__ float DISSF = (float)((1.0/917.0 - 1.0/1000.0)/3.34e5);

// x^1.25 for x > 0
__device__ __forceinline__ float pow125(float x) {
    return x * sqrtf(sqrtf(x));
}

// ---------------------------------------------------------------------
// Kernel 1: per-link fields  grad[l], chan_q[l], sheet_q[l]
// ---------------------------------------------------------------------
__global__ void link_fields_kernel(const float* __restrict__ pot,
                                   const float* __restrict__ csz,
                                   const float* __restrict__ sht,
                                   const float* __restrict__ llen,
                                   const int*   __restrict__ head,
                                   const int*   __restrict__ tail,
                                   float* __restrict__ grad,
                                   float* __restrict__ chanq,
                                   float* __restrict__ sheetq,
                                   int L) {
    int l = blockIdx.x * blockDim.x + threadIdx.x;
    if (l >= L) return;
    int h = head[l], t = tail[l];
    float g = (pot[h] - pot[t]) / llen[l];
    grad[l]  = g;
    chanq[l] = -K_CHAN * pow125(csz[l]) * g;
    float st = 0.5f * (sht[h] + sht[t]);
    sheetq[l] = -K_SHEET * pow125(st) * rsqrtf(fabsf(g)) * g;
}

// ---------------------------------------------------------------------
// Kernel 2: TDM zero-fill of the dense output matrix.
// One wave (32 threads) per block. Zero a 32 KB LDS tile once with
// ds_store_b128, escape the buffer into an asm memory fence so DSE
// cannot delete the stores (the compiler does not model
// tensor_store_from_lds as reading LDS), then stream the tile to global
// memory repeatedly via tensor_store_from_lds DMA descriptors.
// CHUNK = 32000 bytes divides N*N*4 = 400,000,000 exactly for N=10000,
// but the kernel also clamps the last chunk for generality.
// ---------------------------------------------------------------------
#define TDM_CHUNK_BYTES 32000

typedef unsigned int u32x4 __attribute__((ext_vector_type(4)));
typedef int          i32x8 __attribute__((ext_vector_type(8)));
typedef int          i32x4 __attribute__((ext_vector_type(4)));

__global__ __launch_bounds__(32)
void tdm_zero_fill_kernel(float* __restrict__ out,
                          unsigned long long totalBytes,
                          unsigned int numChunks) {
    __shared__ __align__(16) float zbuf[TDM_CHUNK_BYTES / 4];

    // zero the LDS tile (32 lanes, 16B ds_store_b128 per store)
    float4 z4 = make_float4(0.f, 0.f, 0.f, 0.f);
    for (int i = threadIdx.x; i < TDM_CHUNK_BYTES / 16; i += 32) {
        ((float4*)zbuf)[i] = z4;
    }
    // escape zbuf so the stores above are observable by the TDM engine
    // (blocks dead-store elimination), then drain LDS writes
    asm volatile("" : : "r"(&zbuf[0]) : "memory");
    __syncthreads();
    asm volatile("s_wait_dscnt 0\n\ts_wait_storecnt 0" ::: "memory");

    unsigned long long base = (unsigned long long)(uintptr_t)out;

    for (unsigned int c = blockIdx.x; c < numChunks; c += gridDim.x) {
        unsigned long long off   = (unsigned long long)c * TDM_CHUNK_BYTES;
        unsigned long long bytes = totalBytes - off;
        if (bytes > TDM_CHUNK_BYTES) bytes = TDM_CHUNK_BYTES;
        unsigned int elems8 = (unsigned int)(bytes >> 3);   // 8-byte elements
        unsigned long long gaddr = base + off;

        // ---- D# group 0: count=1, lds_addr=0, global_addr, type=2 ----
        u32x4 g0;
        g0[0] = 1u;                                   // count=1, user descriptor
        g0[1] = 0u;                                   // lds_addr (zbuf at LDS offset 0)
        g0[2] = (unsigned int)(gaddr & 0xFFFFFFFFull);        // global_addr[31:0]
        g0[3] = (unsigned int)((gaddr >> 32) & 0x1FFFFFFull)  // global_addr[56:32]
              | (2u << 30);                                    // type = 2 ("image")

        // ---- D# group 1: data_size=8B, 1-D tensor/tile of elems8 ----
        i32x8 g1;
        g1[0] = (int)(3u << 16);                      // workgroup_mask=0, data_size=3 (8B)
        g1[1] = (int)((elems8 & 0xFFFFu) << 16);      // tensor_dim0[15:0]
        g1[2] = (int)(elems8 >> 16);                  // tensor_dim0[31:16], tensor_dim1=0
        g1[3] = (int)((elems8 & 0xFFFFu) << 16);      // tile_dim0
        g1[4] = 0;                                    // tile_dim1=0, tile_dim2=0
        g1[5] = (int)elems8;                          // tensor_dim0_stride[31:0]
        g1[6] = 0;
        g1[7] = 0;

        i32x4 g2 = {0, 0, 0, 0};
        i32x4 g3 = {0, 0, 0, 0};
        i32x8 g4 = {0, 0, 0, 0, 0, 0, 0, 0};

        // amdgpu-toolchain (clang-23) 6-arg form
        __builtin_amdgcn_tensor_store_from_lds(g0, g1, g2, g3, g4, 0);
    }

    __builtin_amdgcn_s_wait_tensorcnt(0);
}

// ---------------------------------------------------------------------
// Kernel 3: scatter the <=5 nonzeros per row (runs after zero-fill).
// Row i only receives contributions from node i itself, so rows are
// fully independent. Boundary rows become identity rows.
// ---------------------------------------------------------------------
__global__ void row_assemble_kernel(const float* __restrict__ csz,
                                    const float* __restrict__ sht,
                                    const float* __restrict__ flen,
                                    const float* __restrict__ llen,
                                    const int*   __restrict__ adj,
                                    const int*   __restrict__ lnk,
                                    const int*   __restrict__ fal,
                                    const int*   __restrict__ inout,
                                    const float* __restrict__ grad,
                                    const float* __restrict__ chanq,
                                    const float* __restrict__ sheetq,
                                    float* __restrict__ M,
                                    int N) {
    int i = blockIdx.x * blockDim.x + threadIdx.x;
    if (i >= N) return;

    size_t row = (size_t)i * (size_t)N;

    if (inout[i] == 1) {            // Dirichlet row -> identity
        M[row + i] = 1.0f;
        return;
    }

    float csi = csz[i];
    float sti = sht[i];
    float diag = 0.0f;

    #pragma unroll
    for (int k = 0; k < 4; ++k) {
        int a = adj[i * 4 + k];
        if (a < 0) continue;        // interior nodes have all 4, but stay faithful
        int l = lnk[i * 4 + k];

        float fl = flen[fal[l]];
        float ll = llen[l];
        float fol = fl / ll;

        float cs = 0.5f * (csi + csz[a]);   // NOTE: link field indexed by node id,
        float st = 0.5f * (sti + sht[a]);   //       reproduced faithfully
        float g  = grad[l];

        float sheet_flux = -K_SHEET * pow125(st) * rsqrtf(fabsf(g)) * fol;
        float chan_flux  = -K_CHAN  * pow125(cs) * fol;
        float ch_diss    = fabsf(DISSF * chanq[l]  * fl);
        float sh_diss    = fabsf(DISSF * sheetq[l] * CAV_SPACE * fl);

        float term = sheet_flux + chan_flux + ch_diss + sh_diss;
        M[row + a] = -term;         // off-diagonal
        diag += term;
    }
    M[row + i] = diag;              // diagonal
}

// ---------------------------------------------------------------------
// Launcher
// ---------------------------------------------------------------------
extern "C" void kernel_launch(void* const* d_in, const int* in_sizes, int n_in,
                              void* d_out, int out_size, void* d_ws, size_t ws_size,
                              hipStream_t stream) {
    const float* pot  = (const float*)d_in[0];   // previous_potential [N]
    const float* csz  = (const float*)d_in[1];   // channel_size       [L]
    const float* sht  = (const float*)d_in[2];   // sheet_thickness    [N]
    const float* flen = (const float*)d_in[3];   // length_of_face     [L]
    const float* llen = (const float*)d_in[4];   // length_of_link     [L]
    const int*   adj  = (const int*)  d_in[5];   // adjacent_nodes     [N,4]
    const int*   lnk  = (const int*)  d_in[6];   // links_at_node      [N,4]
    const int*   fal  = (const int*)  d_in[7];   // face_at_link       [L]
    const int*   head = (const int*)  d_in[8];   // link_head          [L]
    const int*   tail = (const int*)  d_in[9];   // link_tail          [L]
    const int*   inout= (const int*)  d_in[10];  // inflow_outflow     [N]

    int N = in_sizes[0];
    int L = in_sizes[1];

    float* M = (float*)d_out;
    float* grad   = (float*)d_ws;
    float* chanq  = grad  + L;
    float* sheetq = chanq + L;

    // 1) per-link fields
    link_fields_kernel<<<(L + 255) / 256, 256, 0, stream>>>(
        pot, csz, sht, llen, head, tail, grad, chanq, sheetq, L);

    // 2) TDM zero-fill of the dense matrix (400 MB streaming store)
    unsigned long long totalBytes = (unsigned long long)out_size * sizeof(float);
    unsigned int numChunks =
        (unsigned int)((totalBytes + TDM_CHUNK_BYTES - 1) / TDM_CHUNK_BYTES);
    unsigned int zblocks = numChunks < 1024u ? numChunks : 1024u;
    tdm_zero_fill_kernel<<<zblocks, 32, 0, stream>>>(M, totalBytes, numChunks);

    // 3) scatter nonzeros (stream-ordered after the zero-fill)
    row_assemble_kernel<<<(N + 255) / 256, 256, 0, stream>>>(
        csz, sht, flen, llen, adj, lnk, fal, inout,
        grad, chanq, sheetq, M, N);
}